// MultiHeadAttentionLayer_29824252903963
// MI455X (gfx1250) — compile-verified
//
#include <hip/hip_runtime.h>

#define IN_DIM   128
#define QKV_DIM  128   // NUM_HEADS * OUT_DIM
#define NUM_HEADS 8

typedef __attribute__((ext_vector_type(2))) float v2f;
typedef __attribute__((ext_vector_type(8))) float v8f;

// ---------------- zero accumulators (out + z) ----------------
__global__ void gat_zero_kernel(float* __restrict__ out, float* __restrict__ z,
                                int nOut, int nZ) {
    int i = blockIdx.x * blockDim.x + threadIdx.x;
    if (i < nOut) out[i] = 0.0f;
    if (i < nZ)   z[i]   = 0.0f;
}

// ---------------- Q/K/V projection via V_WMMA_F32_16X16X4_F32 ----------------
// grid.x = row tiles (N/16), grid.y = 0..2 selects (Q,K,V); block = 256 = 8 waves,
// wave w computes the 16x16 output tile at columns [16w, 16w+16).
__global__ __launch_bounds__(256) void gat_proj_kernel(
    const float* __restrict__ h,
    const float* __restrict__ WQ, const float* __restrict__ bQ,
    const float* __restrict__ WK, const float* __restrict__ bK,
    const float* __restrict__ WV, const float* __restrict__ bV,
    float* __restrict__ Qp, float* __restrict__ Kp, float* __restrict__ Vp)
{
    const float* W;  const float* bias;  float* Y;
    if (blockIdx.y == 0)      { W = WQ; bias = bQ; Y = Qp; }
    else if (blockIdx.y == 1) { W = WK; bias = bK; Y = Kp; }
    else                      { W = WV; bias = bV; Y = Vp; }

    const int lane    = threadIdx.x & 31;
    const int colBase = (threadIdx.x >> 5) * 16;   // wave id -> column tile
    const int m       = lane & 15;                 // row (A) / col (B) within tile
    const int hi      = lane >> 4;                 // 0: K pair {0,1}, 1: K pair {2,3}
    const int kHalf   = hi * 2;
    const long rowBase = (long)blockIdx.x * 16;
    const float* hRow  = h + (rowBase + m) * IN_DIM;

    v8f acc = {};
    #pragma unroll
    for (int k0 = 0; k0 < IN_DIM; k0 += 4) {
        // A 16x4 f32: lane m holds h[row, k0+kHalf], h[row, k0+kHalf+1] (8B load)
        v2f a = *(const v2f*)(hRow + k0 + kHalf);
        // B 4x16 f32: lane holds W[k0+kHalf, col], W[k0+kHalf+1, col]
        v2f b;
        b.x = W[(k0 + kHalf)     * QKV_DIM + colBase + m];
        b.y = W[(k0 + kHalf + 1) * QKV_DIM + colBase + m];
        acc = __builtin_amdgcn_wmma_f32_16x16x4_f32(
            /*neg_a=*/false, a, /*neg_b=*/false, b,
            /*c_mod=*/(short)0, acc, /*reuse_a=*/false, /*reuse_b=*/false);
    }

    const float bv = bias[colBase + m];
    #pragma unroll
    for (int r = 0; r < 8; ++r) {
        int M = r + hi * 8;   // C/D layout: VGPR r, lanes 16-31 hold M+8
        Y[(rowBase + M) * QKV_DIM + colBase + m] = acc[r] + bv;
    }
}

// ---------------- wave-per-edge attention + scatter ----------------
// 32 lanes cover 128 dims (float4/lane); 4 lanes per head.
__global__ __launch_bounds__(256) void gat_edge_kernel(
    const float* __restrict__ Qp, const float* __restrict__ Kp,
    const float* __restrict__ Vp,
    const int* __restrict__ src, const int* __restrict__ dst,
    float* __restrict__ out, float* __restrict__ z, int nEdges)
{
    const int wave = (int)((blockIdx.x * (unsigned)blockDim.x + threadIdx.x) >> 5);
    const int lane = threadIdx.x & 31;
    if (wave >= nEdges) return;

    const int s = src[wave];
    const int d = dst[wave];

    const float4 kv = ((const float4*)(Kp + (long)s * QKV_DIM))[lane];
    const float4 qv = ((const float4*)(Qp + (long)d * QKV_DIM))[lane];

    float part = kv.x * qv.x + kv.y * qv.y + kv.z * qv.z + kv.w * qv.w;
    // reduce across the 4 lanes of this head group (wave32 shuffles)
    part += __shfl_xor(part, 1, 32);
    part += __shfl_xor(part, 2, 32);

    float sc = part * 0.25f;                         // 1/sqrt(OUT_DIM=16)
    sc = fminf(fmaxf(sc, -5.0f), 5.0f);
    const float e = expf(sc);

    const float4 vv = ((const float4*)(Vp + (long)s * QKV_DIM))[lane];
    float* o = out + (long)d * QKV_DIM + lane * 4;
    atomicAdd(o + 0, vv.x * e);
    atomicAdd(o + 1, vv.y * e);
    atomicAdd(o + 2, vv.z * e);
    atomicAdd(o + 3, vv.w * e);
    if ((lane & 3) == 0)
        atomicAdd(z + (long)d * NUM_HEADS + (lane >> 2), e);
}

// ---------------- normalize ----------------
__global__ void gat_norm_kernel(float* __restrict__ out, const float* __restrict__ z,
                                int total) {
    int i = blockIdx.x * blockDim.x + threadIdx.x;
    if (i >= total) return;
    int n  = i >> 7;          // / 128
    int hh = (i >> 4) & 7;    // head index
    out[i] = out[i] / (z[n * NUM_HEADS + hh] + 1e-6f);
}

extern "C" void kernel_launch(void* const* d_in, const int* in_sizes, int n_in,
                              void* d_out, int out_size, void* d_ws, size_t ws_size,
                              hipStream_t stream) {
    const float* h   = (const float*)d_in[0];
    const int*   src = (const int*)  d_in[1];
    const int*   dst = (const int*)  d_in[2];
    const float* WQ  = (const float*)d_in[3];
    const float* bQ  = (const float*)d_in[4];
    const float* WK  = (const float*)d_in[5];
    const float* bK  = (const float*)d_in[6];
    const float* WV  = (const float*)d_in[7];
    const float* bV  = (const float*)d_in[8];

    const int N = in_sizes[0] / IN_DIM;   // 50000
    const int E = in_sizes[1];            // 800000

    float* out = (float*)d_out;
    float* Qp  = (float*)d_ws;
    float* Kp  = Qp + (size_t)N * QKV_DIM;
    float* Vp  = Kp + (size_t)N * QKV_DIM;
    float* z   = Vp + (size_t)N * QKV_DIM;

    const int nOut = N * QKV_DIM;         // 6.4M
    const int nZ   = N * NUM_HEADS;

    // 1) zero accumulators (harness does not re-zero d_out between replays)
    gat_zero_kernel<<<(nOut + 255) / 256, 256, 0, stream>>>(out, z, nOut, nZ);

    // 2) Q/K/V projections: grid (N/16 row tiles, 3 matrices), 8 waves/block
    dim3 pg(N / 16, 3, 1);
    gat_proj_kernel<<<pg, 256, 0, stream>>>(h, WQ, bQ, WK, bK, WV, bV, Qp, Kp, Vp);

    // 3) edge attention + scatter: one wave per edge, 8 waves/block
    gat_edge_kernel<<<(E + 7) / 8, 256, 0, stream>>>(Qp, Kp, Vp, src, dst, out, z, E);

    // 4) normalize by z
    gat_norm_kernel<<<(nOut + 255) / 256, 256, 0, stream>>>(out, z, nOut);
}